// SE3Comp_2954937499888
// MI455X (gfx1250) — compile-verified
//
#include <hip/hip_runtime.h>
#include <cstdint>

#define TILE 256u

// ---- CDNA5 async global<->LDS DMA helpers (ASYNCcnt-tracked) ----------------
// LDS operand is a byte offset relative to the workgroup LDS base; the low 32
// bits of a flat shared-memory address are exactly that offset.
__device__ __forceinline__ uint32_t lds_off_u32(const void* p) {
    return (uint32_t)(uintptr_t)p;
}

__device__ __forceinline__ void async_g2l_b128(uint32_t ldsOff, uint32_t gOff, const void* base) {
    asm volatile("global_load_async_to_lds_b128 %0, %1, %2 th:TH_LOAD_NT"
                 :: "v"(ldsOff), "v"(gOff), "s"(base) : "memory");
}
__device__ __forceinline__ void async_g2l_b32(uint32_t ldsOff, uint32_t gOff, const void* base) {
    asm volatile("global_load_async_to_lds_b32 %0, %1, %2 th:TH_LOAD_NT"
                 :: "v"(ldsOff), "v"(gOff), "s"(base) : "memory");
}
__device__ __forceinline__ void async_l2g_b128(uint32_t gOff, uint32_t ldsOff, const void* base) {
    asm volatile("global_store_async_from_lds_b128 %0, %1, %2 th:TH_STORE_NT"
                 :: "v"(gOff), "v"(ldsOff), "s"(base) : "memory");
}
__device__ __forceinline__ void async_l2g_b32(uint32_t gOff, uint32_t ldsOff, const void* base) {
    asm volatile("global_store_async_from_lds_b32 %0, %1, %2 th:TH_STORE_NT"
                 :: "v"(gOff), "v"(ldsOff), "s"(base) : "memory");
}
__device__ __forceinline__ void wait_async0() {
    asm volatile("s_wait_asynccnt 0x0" ::: "memory");
}

// -----------------------------------------------------------------------------
__global__ void __launch_bounds__(TILE)
se3comp_kernel(const float* __restrict__ Tg, const float* __restrict__ xi,
               float* __restrict__ out, int n) {
    __shared__ __align__(16) float sTg[TILE * 7];   // 7168 B
    __shared__ __align__(16) float sXi[TILE * 6];   // 6144 B
    __shared__ __align__(16) float sOut[TILE * 7];  // 7168 B

    const uint32_t tid = threadIdx.x;
    const uint32_t e0  = blockIdx.x * TILE;                 // first element of tile
    const uint32_t rem = (uint32_t)n - e0;
    const uint32_t cnt = rem < TILE ? rem : TILE;           // elements in this tile

    const uint32_t ldsTg  = lds_off_u32(sTg);
    const uint32_t ldsXi  = lds_off_u32(sXi);
    const uint32_t ldsOut = lds_off_u32(sOut);

    const uint32_t tgBase  = e0 * 28u;   // byte offsets (block base is 16B aligned)
    const uint32_t xiBase  = e0 * 24u;
    const uint32_t tgBytes = cnt * 28u;
    const uint32_t xiBytes = cnt * 24u;

    // ---- Stage inputs: coalesced async DMA global -> LDS --------------------
    for (uint32_t i = tid; i < (tgBytes >> 4); i += TILE)
        async_g2l_b128(ldsTg + (i << 4), tgBase + (i << 4), Tg);
    for (uint32_t i = tid; i < (xiBytes >> 4); i += TILE)
        async_g2l_b128(ldsXi + (i << 4), xiBase + (i << 4), xi);
    {   // 4-byte tail (only for non-multiple-of-TILE tiles)
        uint32_t d = tgBytes & ~15u;
        for (uint32_t i = tid; i < ((tgBytes & 15u) >> 2); i += TILE)
            async_g2l_b32(ldsTg + d + (i << 2), tgBase + d + (i << 2), Tg);
        d = xiBytes & ~15u;
        for (uint32_t i = tid; i < ((xiBytes & 15u) >> 2); i += TILE)
            async_g2l_b32(ldsXi + d + (i << 2), xiBase + d + (i << 2), xi);
    }
    wait_async0();
    __syncthreads();

    // ---- Per-element SE(3) composition (fp32 VALU) --------------------------
    if (tid < cnt) {
        const float* tg = &sTg[tid * 7u];
        const float tx = tg[0], ty = tg[1], tz = tg[2];
        const float qw = tg[3], qx = tg[4], qy = tg[5], qz = tg[6];
        const float* xp = &sXi[tid * 6u];
        const float rx = xp[0], ry = xp[1], rz = xp[2];
        const float ox = xp[3], oy = xp[4], oz = xp[5];

        // so3_RV: R = I + A*S + B*S^2 ; V = I + B*S + C*S^2
        const float oxx = ox * ox, oyy = oy * oy, ozz = oz * oz;
        const float oxy = ox * oy, oxz = ox * oz, oyz = oy * oz;
        const float th2 = oxx + oyy + ozz;
        const float th  = sqrtf(th2);
        const float th3 = th2 * th;
        const float sn = __sinf(th), cs = __cosf(th);
        const float A = th > 0.f ? sn / th : 0.f;
        const float th4 = th2 * th2, th6 = th4 * th2, th8 = th4 * th4;
        const float Bex = (1.f - cs) / (th2 > 0.f ? th2 : 1.f);
        const float Bty = 0.5f + th4 * (1.f / 720.f) + th8 * (1.f / 3628800.f)
                        - th2 * (1.f / 24.f) - th6 * (1.f / 40320.f);
        const float Bc = th2 > 0.1f ? Bex : (th2 < 1e-6f ? 0.f : Bty);
        const float Cex = (th - sn) / (th3 > 0.f ? th3 : 1.f);
        const float Cty = (1.f / 6.f) + th4 * (1.f / 5040.f) + th8 * (1.f / 39916800.f)
                        - th2 * (1.f / 120.f) - th6 * (1.f / 362880.f);
        const float Cc = th3 > 0.1f ? Cex : (th2 < 1e-6f ? 0.f : Cty);

        const float S2_00 = -(oyy + ozz), S2_11 = -(oxx + ozz), S2_22 = -(oxx + oyy);
        const float r00 = 1.f + Bc * S2_00;
        const float r01 = -A * oz + Bc * oxy;
        const float r02 =  A * oy + Bc * oxz;
        const float r10 =  A * oz + Bc * oxy;
        const float r11 = 1.f + Bc * S2_11;
        const float r12 = -A * ox + Bc * oyz;
        const float r20 = -A * oy + Bc * oxz;
        const float r21 =  A * ox + Bc * oyz;
        const float r22 = 1.f + Bc * S2_22;

        const float v00 = 1.f + Cc * S2_00;
        const float v01 = -Bc * oz + Cc * oxy;
        const float v02 =  Bc * oy + Cc * oxz;
        const float v10 =  Bc * oz + Cc * oxy;
        const float v11 = 1.f + Cc * S2_11;
        const float v12 = -Bc * ox + Cc * oyz;
        const float v20 = -Bc * oy + Cc * oxz;
        const float v21 =  Bc * ox + Cc * oyz;
        const float v22 = 1.f + Cc * S2_22;

        const float txi0 = v00 * rx + v01 * ry + v02 * rz;
        const float txi1 = v10 * rx + v11 * ry + v12 * rz;
        const float txi2 = v20 * rx + v21 * ry + v22 * rz;

        // Rg from quaternion (qw,qx,qy,qz)
        const float g00 = 1.f - 2.f * (qy * qy + qz * qz);
        const float g01 = 2.f * (qx * qy - qz * qw);
        const float g02 = 2.f * (qx * qz + qy * qw);
        const float g10 = 2.f * (qx * qy + qz * qw);
        const float g11 = 1.f - 2.f * (qx * qx + qz * qz);
        const float g12 = 2.f * (qy * qz - qx * qw);
        const float g20 = 2.f * (qx * qz - qy * qw);
        const float g21 = 2.f * (qy * qz + qx * qw);
        const float g22 = 1.f - 2.f * (qx * qx + qy * qy);

        // Rc = R @ Rg
        const float m00 = r00 * g00 + r01 * g10 + r02 * g20;
        const float m01 = r00 * g01 + r01 * g11 + r02 * g21;
        const float m02 = r00 * g02 + r01 * g12 + r02 * g22;
        const float m10 = r10 * g00 + r11 * g10 + r12 * g20;
        const float m11 = r10 * g01 + r11 * g11 + r12 * g21;
        const float m12 = r10 * g02 + r11 * g12 + r12 * g22;
        const float m20 = r20 * g00 + r21 * g10 + r22 * g20;
        const float m21 = r20 * g01 + r21 * g11 + r22 * g21;
        const float m22 = r20 * g02 + r21 * g12 + r22 * g22;

        // tc = R*tg + t_xi
        const float tc0 = r00 * tx + r01 * ty + r02 * tz + txi0;
        const float tc1 = r10 * tx + r11 * ty + r12 * tz + txi1;
        const float tc2 = r20 * tx + r21 * ty + r22 * tz + txi2;

        // m_to_r7: branchless quaternion extraction
        const bool cond1  = m22 < 0.f;
        const bool cond1a = m00 > m11;
        const bool cond2  = m00 < -m11;
        const float t0 = 1.f + m00 - m11 - m22;
        const float t1 = 1.f - m00 + m11 - m22;
        const float t2 = 1.f - m00 - m11 + m22;
        const float t3 = 1.f + m00 + m11 + m22;
        auto sel = [&](float a, float b, float c, float d) {
            return cond1 ? (cond1a ? a : b) : (cond2 ? c : d);
        };
        const float tt = sel(t0, t1, t2, t3);
        float uw = sel(m21 - m12, m02 - m20, m10 - m01, t3);
        float ux = sel(t0, m01 + m10, m20 + m02, m21 - m12);
        float uy = sel(m01 + m10, t1, m12 + m21, m02 - m20);
        float uz = sel(m20 + m02, m12 + m21, t2, m10 - m01);
        const float sc = 0.5f / sqrtf(tt);
        uw *= sc; ux *= sc; uy *= sc; uz *= sc;
        const float sg = uw < 0.f ? -1.f : 1.f;
        uw *= sg; ux *= sg; uy *= sg; uz *= sg;

        float* o = &sOut[tid * 7u];
        o[0] = tc0; o[1] = tc1; o[2] = tc2;
        o[3] = uw;  o[4] = ux;  o[5] = uy;  o[6] = uz;
    }
    __syncthreads();   // DS writes visible before the async store engine reads LDS

    // ---- Drain results: coalesced async DMA LDS -> global -------------------
    const uint32_t outBase  = e0 * 28u;
    const uint32_t outBytes = cnt * 28u;
    for (uint32_t i = tid; i < (outBytes >> 4); i += TILE)
        async_l2g_b128(outBase + (i << 4), ldsOut + (i << 4), out);
    {
        const uint32_t d = outBytes & ~15u;
        for (uint32_t i = tid; i < ((outBytes & 15u) >> 2); i += TILE)
            async_l2g_b32(outBase + d + (i << 2), ldsOut + d + (i << 2), out);
    }
    wait_async0();  // s_endpgm's implicit wait-idle also covers this
}

extern "C" void kernel_launch(void* const* d_in, const int* in_sizes, int n_in,
                              void* d_out, int out_size, void* d_ws, size_t ws_size,
                              hipStream_t stream) {
    const float* Tg = (const float*)d_in[0];   // B x 7 x 1 fp32
    const float* xi = (const float*)d_in[1];   // B x 6 x 1 fp32
    float* out = (float*)d_out;                // B x 7 x 1 fp32

    const int n = in_sizes[0] / 7;             // B = 2,097,152
    const int blocks = (n + (int)TILE - 1) / (int)TILE;
    se3comp_kernel<<<blocks, TILE, 0, stream>>>(Tg, xi, out, n);
}